// CovFea_1dBV_11699490914564
// MI455X (gfx1250) — compile-verified
//
#include <hip/hip_runtime.h>

typedef float v2f __attribute__((ext_vector_type(2)));
typedef float v8f __attribute__((ext_vector_type(8)));

static constexpr int kF = 4096;          // features (fastest-moving dim in memory)
static constexpr int kN = 16384;         // V*B samples per feature
static constexpr int kTilesF = kF / 16;  // 256 feature tiles of 16
static constexpr int kRowChunks = 32;    // row-range splits (occupancy: ~8 waves/SIMD)
static constexpr int kRowsPerWave = kN / kRowChunks;  // 512
static constexpr int kWavesPerBlock = 8;
static constexpr int kThreads = kWavesPerBlock * 32;  // 256
static constexpr int kTotalWaves = kTilesF * kRowChunks;        // 8192
static constexpr int kBlocks = kTotalWaves / kWavesPerBlock;    // 1024

// --- zero the moment accumulators in workspace (harness poisons d_ws) ---
__global__ void cov_zero_ws(float* __restrict__ s) {
    int i = blockIdx.x * blockDim.x + threadIdx.x;
    if (i < 4 * kF) s[i] = 0.0f;
}

// --- single-pass raw moments s1..s4 per feature, WMMA-accumulated ---
__global__ __launch_bounds__(kThreads)
void cov_moments_kernel(const float* __restrict__ zs, float* __restrict__ S) {
    const int wave = blockIdx.x * kWavesPerBlock + (threadIdx.x >> 5);
    const int lane = threadIdx.x & 31;
    const int ft = wave & (kTilesF - 1);   // feature tile 0..255
    const int rc = wave >> 8;              // row chunk 0..31
    const int f0 = ft * 16;
    const int half = lane >> 4;            // 0: rows r,r+1 ; 1: rows r+2,r+3
    const int l16 = lane & 15;             // feature M within tile (A-matrix lane map)

    const float* p = zs + (size_t)(rc * kRowsPerWave + half * 2) * kF + f0 + l16;

    const v2f ones = {1.0f, 1.0f};         // B = all-ones 4x16 -> D[m,n] = sum_k A[m,k]
    v8f c1 = {}; v8f c2 = {}; v8f c3 = {}; v8f c4 = {};

    for (int r = 0; r < kRowsPerWave; r += 4) {
        float x0 = p[0];                   // row r + half*2
        float x1 = p[kF];                  // row r + half*2 + 1
        __builtin_prefetch(p + 16 * kF, 0, 3);   // gfx1250 global_prefetch, WGP scope
        p += 4 * kF;

        v2f a  = {x0, x1};                 // A tile: 16 features x 4 rows (fp32)
        v2f a2 = a * a;                    // v_pk_mul_f32
        v2f a3 = a2 * a;
        v2f a4 = a2 * a2;

        // D = A x ones + C : fp32 row-sums for 16 features per instruction
        c1 = __builtin_amdgcn_wmma_f32_16x16x4_f32(false, a,  false, ones, (short)0, c1, false, false);
        c2 = __builtin_amdgcn_wmma_f32_16x16x4_f32(false, a2, false, ones, (short)0, c2, false, false);
        c3 = __builtin_amdgcn_wmma_f32_16x16x4_f32(false, a3, false, ones, (short)0, c3, false, false);
        c4 = __builtin_amdgcn_wmma_f32_16x16x4_f32(false, a4, false, ones, (short)0, c4, false, false);
    }

    // D layout: VGPR r holds M=r (lanes 0-15) and M=8+r (lanes 16-31); every
    // column N is identical, so lane 0 / lane 16 each own 8 feature sums.
    if (l16 == 0) {
        const int fb = f0 + half * 8;
#pragma unroll
        for (int r = 0; r < 8; ++r) {
            atomicAdd(&S[0 * kF + fb + r], c1[r]);
            atomicAdd(&S[1 * kF + fb + r], c2[r]);
            atomicAdd(&S[2 * kF + fb + r], c3[r]);
            atomicAdd(&S[3 * kF + fb + r], c4[r]);
        }
    }
}

// --- centered-moment algebra + mean over features -> scalar ---
__global__ void cov_finalize_kernel(const float* __restrict__ S, float* __restrict__ out) {
    __shared__ float red[256];
    const float N = (float)kN;
    float acc = 0.0f;
    for (int f = threadIdx.x; f < kF; f += 256) {
        float s1 = S[0 * kF + f];
        float s2 = S[1 * kF + f];
        float s3 = S[2 * kF + f];
        float s4 = S[3 * kF + f];
        float m   = s1 / N;
        float m2  = m * m;
        float sc2 = s2 - s1 * m;                                    // sum (z-m)^2
        float sc4 = s4 - 4.0f * m * s3 + 6.0f * m2 * s2 - 3.0f * N * m2 * m2;  // sum (z-m)^4
        float nm1 = N - 1.0f;
        acc += (sc2 * sc2 - sc4) / (nm1 * nm1 * nm1);
    }
    red[threadIdx.x] = acc;
    __syncthreads();
    for (int s = 128; s > 0; s >>= 1) {
        if ((int)threadIdx.x < s) red[threadIdx.x] += red[threadIdx.x + s];
        __syncthreads();
    }
    if (threadIdx.x == 0) out[0] = red[0] / (float)kF;
}

extern "C" void kernel_launch(void* const* d_in, const int* in_sizes, int n_in,
                              void* d_out, int out_size, void* d_ws, size_t ws_size,
                              hipStream_t stream) {
    (void)in_sizes; (void)n_in; (void)out_size; (void)ws_size;
    const float* zs = (const float*)d_in[0];
    float* S = (float*)d_ws;            // 4 * 4096 floats = 64 KiB of scratch
    float* out = (float*)d_out;

    cov_zero_ws<<<(4 * kF + 255) / 256, 256, 0, stream>>>(S);
    cov_moments_kernel<<<kBlocks, kThreads, 0, stream>>>(zs, S);
    cov_finalize_kernel<<<1, 256, 0, stream>>>(S, out);
}